// DepthwiseConv3D_7980049236460
// MI455X (gfx1250) — compile-verified
//
#include <hip/hip_runtime.h>

// Depthwise 3D conv 3x3x3, SAME, NDHWC, N=4 D=16 H=W=112 C=64, fp32.
// Memory-bound (AI ~6.7 flop/B): optimize the data path with CDNA5
// async global->LDS staging (ASYNCcnt) + b128 coalesced channel-vectors.

#define C_CH    64
#define TW      16          // output W pixels per block
#define TH      4           // output H rows per block
#define HW      (TW + 2)    // 18 staged pixels in W (halo)
#define HH      (TH + 2)    // 6 staged rows in H (halo)
#define NTAP    27
#define THREADS 256

#define IN_ELEMS (3 * HH * HW * 16)   // 5184 float4 (3 d-slices)
#define W_ELEMS  (NTAP * 16)          // 432  float4

typedef int v4i __attribute__((ext_vector_type(4)));
typedef __attribute__((address_space(1))) v4i gv4i;   // global int4
typedef __attribute__((address_space(3))) v4i lv4i;   // LDS int4

__device__ __forceinline__ void async_copy_b128(const void* gptr, void* lptr_generic) {
  // Low 32 bits of a generic LDS pointer are the LDS byte offset (ISA 10.2).
  unsigned int lds_off = (unsigned int)(unsigned long long)lptr_generic;
#if __has_builtin(__builtin_amdgcn_global_load_async_to_lds_b128)
  gv4i* gp = (gv4i*)(unsigned long long)gptr;
  lv4i* lp = (lv4i*)(unsigned long long)lds_off;
  __builtin_amdgcn_global_load_async_to_lds_b128(gp, lp, 0, 0);
#else
  asm volatile("global_load_async_to_lds_b128 %0, %1, off"
               :: "v"(lds_off), "v"(gptr) : "memory");
#endif
}

__device__ __forceinline__ void wait_async0() {
#if __has_builtin(__builtin_amdgcn_s_wait_asynccnt)
  __builtin_amdgcn_s_wait_asynccnt(0);
#else
  asm volatile("s_wait_asynccnt 0" ::: "memory");
#endif
}

__global__ __launch_bounds__(THREADS)
void dwconv3d_f32_kernel(const float* __restrict__ x,
                         const float* __restrict__ w,
                         float* __restrict__ out) {
  __shared__ float4 s_in[IN_ELEMS];   // [dd(3)][hh(6)][px(18)][cg(16)]
  __shared__ float4 s_w[W_ELEMS];     // [tap(27)][cg(16)]

  const int tid = threadIdx.x;
  const int wt  = blockIdx.x;         // 0..6   (112/16)
  const int ht  = blockIdx.y;         // 0..27  (112/4)
  const int nd  = blockIdx.z;         // 0..63  (4*16)
  const int d   = nd & 15;
  const int n   = nd >> 4;
  const int h0  = ht * TH;
  const int w0  = wt * TW;

  // ---- stage all 27x64 weights (tiny, L2-resident after first block) ----
  for (int e = tid; e < W_ELEMS; e += THREADS) {
    // float index = tap*64 + cg*4 == e*4
    async_copy_b128(w + (size_t)e * 4, &s_w[e]);
  }

  // ---- stage input tile (with halo) via async b128; zero-fill padding ----
  for (int e = tid; e < IN_ELEMS; e += THREADS) {
    const int cg = e & 15;
    const int t  = e >> 4;      // 0..323
    const int px = t % HW;      // 0..17
    const int t2 = t / HW;      // 0..17
    const int hh = t2 % HH;     // 0..5
    const int dd = t2 / HH;     // 0..2
    const int gd = d  + dd - 1;
    const int gh = h0 + hh - 1;
    const int gw = w0 + px - 1;
    if ((unsigned)gd < 16u && (unsigned)gh < 112u && (unsigned)gw < 112u) {
      const size_t gidx =
          (((size_t)(n * 16 + gd) * 112 + gh) * 112 + gw) * C_CH + cg * 4;
      async_copy_b128(x + gidx, &s_in[e]);
    } else {
      s_in[e] = make_float4(0.f, 0.f, 0.f, 0.f);   // SAME zero padding
    }
  }

  wait_async0();        // ASYNCcnt drained for this wave
  __syncthreads();      // DScnt + cross-wave visibility

  // ---- compute: each thread = 4 channels x 1 W-pixel x TH rows ----
  const int cg = tid & 15;   // channel group (4 floats)
  const int px = tid >> 4;   // 0..15 pixel in W tile

  float4 acc[TH];
#pragma unroll
  for (int r = 0; r < TH; ++r) acc[r] = make_float4(0.f, 0.f, 0.f, 0.f);

#pragma unroll
  for (int kd = 0; kd < 3; ++kd) {
#pragma unroll
    for (int kh = 0; kh < 3; ++kh) {
#pragma unroll
      for (int kw = 0; kw < 3; ++kw) {
        const float4 wv = s_w[((kd * 3 + kh) * 3 + kw) * 16 + cg];
#pragma unroll
        for (int r = 0; r < TH; ++r) {
          const float4 xv =
              s_in[((kd * HH + (r + kh)) * HW + (px + kw)) * 16 + cg];
          acc[r].x = fmaf(xv.x, wv.x, acc[r].x);
          acc[r].y = fmaf(xv.y, wv.y, acc[r].y);
          acc[r].z = fmaf(xv.z, wv.z, acc[r].z);
          acc[r].w = fmaf(xv.w, wv.w, acc[r].w);
        }
      }
    }
  }

  // ---- coalesced b128 stores ----
#pragma unroll
  for (int r = 0; r < TH; ++r) {
    const size_t oidx =
        (((size_t)(n * 16 + d) * 112 + (h0 + r)) * 112 + (w0 + px)) * C_CH +
        cg * 4;
    *reinterpret_cast<float4*>(out + oidx) = acc[r];
  }
}

extern "C" void kernel_launch(void* const* d_in, const int* in_sizes, int n_in,
                              void* d_out, int out_size, void* d_ws, size_t ws_size,
                              hipStream_t stream) {
  (void)in_sizes; (void)n_in; (void)out_size; (void)d_ws; (void)ws_size;
  const float* x = (const float*)d_in[0];   // (4,16,112,112,64) f32
  const float* w = (const float*)d_in[1];   // (3,3,3,1,64) f32
  float* out = (float*)d_out;               // (4,16,112,112,64) f32

  dim3 grid(112 / TW, 112 / TH, 4 * 16);    // (7, 28, 64)
  dwconv3d_f32_kernel<<<grid, THREADS, 0, stream>>>(x, w, out);
}